// KWJAHeads_69965017252516
// MI455X (gfx1250) — compile-verified
//
#include <hip/hip_runtime.h>
#include <hip/hip_bf16.h>
#include <math.h>

// Shapes
#define BATCH 1
#define SEQ   128
#define HID   768
#define NL    10
#define NR    2

typedef __attribute__((ext_vector_type(2))) float v2f;
typedef __attribute__((ext_vector_type(8))) float v8f;

// ---------------------------------------------------------------------------
// Phase 0: xa[s,l,r] = sum_h x[s,h] * A[h,r,l]       (A is (H, R, L))
// one thread per (s,l,r): 128*10*2 = 2560 threads
// ---------------------------------------------------------------------------
__global__ __launch_bounds__(256)
void xa_kernel(const float* __restrict__ X, const float* __restrict__ A,
               float* __restrict__ xa) {
    int idx = blockIdx.x * blockDim.x + threadIdx.x;
    if (idx >= SEQ * NL * NR) return;
    int r = idx & 1;
    int l = (idx >> 1) % NL;
    int s = idx / (NL * NR);
    const float* xrow = X + s * HID;
    float sum = 0.0f;
    #pragma unroll 8
    for (int h = 0; h < HID; ++h)
        sum += xrow[h] * A[(h * NR + r) * NL + l];
    xa[idx] = sum;   // layout (s, l, r)
}

// ---------------------------------------------------------------------------
// Phase 1: WMMA f32 GEMM + fused bias + rank-2 LoRA epilogue.
// out[s,l,h] = (X @ W)[s,h] + bias[h] + sum_r xa[s,l,r] * Blora[r,h,l]
// One wave (32 threads) computes a 16x16 (s,h) tile via V_WMMA_F32_16X16X4_F32
// over K=768, then emits all 10 adapter variants from the one accumulator.
// grid = (H/16, SEQ/16), block = 32.
// ---------------------------------------------------------------------------
__global__ __launch_bounds__(32)
void fused_linear_wmma(const float* __restrict__ X,      // (SEQ, HID)
                       const float* __restrict__ W,      // (HID, HID) in-major
                       const float* __restrict__ bias,   // (HID)
                       const float* __restrict__ Blora,  // (R, HID, L)
                       const float* __restrict__ xa,     // (SEQ, L, R)
                       float* __restrict__ out)          // (SEQ, L, HID)
{
    const int hTile = blockIdx.x * 16;
    const int sTile = blockIdx.y * 16;
    const int lane  = threadIdx.x;          // 0..31, full wave, EXEC all-1s
    const int nLoc  = lane & 15;            // M index for A-frag / N for B-frag
    const int kOff  = (lane >> 4) * 2;      // lanes 16-31 hold K+2, K+3

    v8f acc = {};
    const float* arow = X + (sTile + nLoc) * HID;

    for (int k0 = 0; k0 < HID; k0 += 4) {
        const int kk = k0 + kOff;
        // A-frag: row (sTile+nLoc), cols kk, kk+1 (kk even -> 8B aligned)
        v2f a = *(const v2f*)(arow + kk);
        // B-frag: col (hTile+nLoc), rows kk, kk+1
        v2f b;
        b.x = W[kk * HID + (hTile + nLoc)];
        b.y = W[(kk + 1) * HID + (hTile + nLoc)];
        acc = __builtin_amdgcn_wmma_f32_16x16x4_f32(
                  false, a, false, b, (short)0, acc, false, false);
    }

    // Epilogue: C/D layout -> lane holds N = (lane&15); VGPR i holds
    // M = i (lanes 0-15) or M = 8+i (lanes 16-31).
    const int n     = hTile + nLoc;
    const int mBase = (lane < 16) ? 0 : 8;
    const float bv  = bias[n];

    #pragma unroll
    for (int l = 0; l < NL; ++l) {
        const float bl0 = Blora[(0 * HID + n) * NL + l];
        const float bl1 = Blora[(1 * HID + n) * NL + l];
        #pragma unroll
        for (int i = 0; i < 8; ++i) {
            const int s = sTile + mBase + i;
            const float xa0 = xa[(s * NL + l) * NR + 0];
            const float xa1 = xa[(s * NL + l) * NR + 1];
            out[(s * NL + l) * HID + n] = acc[i] + bv + xa0 * bl0 + xa1 * bl1;
        }
    }
}

// ---------------------------------------------------------------------------
// Phase 2: out[l,s,t] = sum_h gelu(src[s,l,h] + tgt[t,l,h]) * cls[h,l]
// Exact GELU (erf). One block per (s,l); src row + classifier column staged
// in LDS; 8 waves stream the 128 target rows; wave32 shuffle reduction.
// grid = (SEQ, NL), block = 256.
// ---------------------------------------------------------------------------
__global__ __launch_bounds__(256)
void pair_gelu_dot(const float* __restrict__ srcOut,   // (SEQ, L, HID)
                   const float* __restrict__ tgtOut,   // (SEQ, L, HID)
                   const float* __restrict__ cls,      // (HID, L)
                   float* __restrict__ out)            // (L, SEQ, SEQ)
{
    const int s = blockIdx.x;
    const int l = blockIdx.y;

    __shared__ float sSrc[HID];
    __shared__ float sCls[HID];
    for (int h = threadIdx.x; h < HID; h += 256) {
        sSrc[h] = srcOut[(s * NL + l) * HID + h];
        sCls[h] = cls[h * NL + l];
    }
    __syncthreads();

    const int wave = threadIdx.x >> 5;
    const int lane = threadIdx.x & 31;
    const float inv_sqrt2 = 0.70710678118654752440f;

    for (int t = wave; t < SEQ; t += 8) {
        const float* trow = tgtOut + (t * NL + l) * HID;
        float sum = 0.0f;
        #pragma unroll 4
        for (int h = lane; h < HID; h += 32) {
            const float v = sSrc[h] + trow[h];
            const float g = 0.5f * v * (1.0f + erff(v * inv_sqrt2));
            sum += g * sCls[h];
        }
        // wave32 butterfly reduction
        #pragma unroll
        for (int off = 16; off > 0; off >>= 1)
            sum += __shfl_xor(sum, off, 32);
        if (lane == 0)
            out[(l * SEQ + s) * SEQ + t] = sum;
    }
}

// ---------------------------------------------------------------------------
// Host-side launcher
// ---------------------------------------------------------------------------
extern "C" void kernel_launch(void* const* d_in, const int* in_sizes, int n_in,
                              void* d_out, int out_size, void* d_ws, size_t ws_size,
                              hipStream_t stream) {
    (void)in_sizes; (void)n_in; (void)out_size; (void)ws_size;

    const float* x     = (const float*)d_in[0];  // (1, 128, 768)
    const float* W_src = (const float*)d_in[1];  // (768, 768)
    const float* b_src = (const float*)d_in[2];  // (768)
    const float* W_tgt = (const float*)d_in[3];
    const float* b_tgt = (const float*)d_in[4];
    const float* A_src = (const float*)d_in[5];  // (768, 2, 10)
    const float* B_src = (const float*)d_in[6];  // (2, 768, 10)
    const float* A_tgt = (const float*)d_in[7];
    const float* B_tgt = (const float*)d_in[8];
    const float* cls   = (const float*)d_in[9];  // (768, 10)

    float* ws     = (float*)d_ws;
    float* srcOut = ws;                                   // 128*10*768
    float* tgtOut = srcOut + SEQ * NL * HID;              // 128*10*768
    float* xaSrc  = tgtOut + SEQ * NL * HID;              // 128*10*2
    float* xaTgt  = xaSrc + SEQ * NL * NR;                // 128*10*2

    // Phase 0: LoRA down-projections
    const int xaThreads = SEQ * NL * NR;                  // 2560
    xa_kernel<<<(xaThreads + 255) / 256, 256, 0, stream>>>(x, A_src, xaSrc);
    xa_kernel<<<(xaThreads + 255) / 256, 256, 0, stream>>>(x, A_tgt, xaTgt);

    // Phase 1: WMMA GEMMs with fused bias + LoRA epilogue
    dim3 g1(HID / 16, SEQ / 16);
    fused_linear_wmma<<<g1, 32, 0, stream>>>(x, W_src, b_src, B_src, xaSrc, srcOut);
    fused_linear_wmma<<<g1, 32, 0, stream>>>(x, W_tgt, b_tgt, B_tgt, xaTgt, tgtOut);

    // Phase 2: pairwise GELU + classifier dot
    dim3 g2(SEQ, NL);
    pair_gelu_dot<<<g2, 256, 0, stream>>>(srcOut, tgtOut, cls, (float*)d_out);
}